// GraphConvolution_14121852469580
// MI455X (gfx1250) — compile-verified
//
#include <hip/hip_runtime.h>

typedef __attribute__((ext_vector_type(2))) float v2f;
typedef __attribute__((ext_vector_type(8))) float v8f;

#define IN_DIM  256
#define OUT_DIM 128
#define LDS_STRIDE 260   // 256 + 4 pad -> row r starts at bank (4*r)%64: conflict-free column reads

// ---------------------------------------------------------------------------
// Kernel 0: out[n][j] = bias[j]  (folds bias in; also required since the
// harness poisons d_out and we accumulate with atomics afterwards)
// ---------------------------------------------------------------------------
__global__ void __launch_bounds__(256) gc_init_bias(const float* __restrict__ bias,
                                                    float* __restrict__ out,
                                                    int total) {
    int i = blockIdx.x * blockDim.x + threadIdx.x;
    if (i < total) out[i] = bias[i & (OUT_DIM - 1)];
}

// ---------------------------------------------------------------------------
// Kernel 1: support = X @ W  via V_WMMA_F32_16X16X4_F32
//   grid.x = nrows/16 (one 16-row M-tile per block), block = 256 (8 waves)
//   wave w computes the 16x16 tile at columns [16w, 16w+16)
// ---------------------------------------------------------------------------
__global__ void __launch_bounds__(256) gc_gemm_wmma(const float* __restrict__ X,
                                                    const float* __restrict__ W,
                                                    float* __restrict__ S) {
    __shared__ float lx[16 * LDS_STRIDE];

    const int tid   = threadIdx.x;
    const int mbase = blockIdx.x * 16;

    // Cooperative, coalesced stage of the 16x256 X tile (float4 per thread-iter)
    for (int i = tid; i < 16 * (IN_DIM / 4); i += 256) {
        int row = i >> 6;       // i / 64
        int c4  = i & 63;       // float4 index within row
        float4 v = ((const float4*)(X + (size_t)(mbase + row) * IN_DIM))[c4];
        float* dst = &lx[row * LDS_STRIDE + c4 * 4];
        dst[0] = v.x; dst[1] = v.y; dst[2] = v.z; dst[3] = v.w;
    }
    __syncthreads();

    const int wave = tid >> 5;
    const int lane = tid & 31;
    const int l16  = lane & 15;
    const int hi   = lane >> 4;     // 0 -> holds K+0,K+1 ; 1 -> holds K+2,K+3
    const int n0   = wave * 16;

    v8f acc = {};
    // A fragment source: row (M=l16), starting K offset per half-wave
    const float* arow = &lx[l16 * LDS_STRIDE + hi * 2];
    // B fragment source: W[k][n0 + l16], k offset per half-wave
    const float* wp = W + (size_t)(hi * 2) * OUT_DIM + n0 + l16;

    #pragma unroll 4
    for (int kt = 0; kt < IN_DIM / 4; ++kt) {
        v2f a, b;
        a.x = arow[kt * 4 + 0];
        a.y = arow[kt * 4 + 1];
        b.x = wp[(size_t)(kt * 4 + 0) * OUT_DIM];
        b.y = wp[(size_t)(kt * 4 + 1) * OUT_DIM];
        // D = A(16x4) * B(4x16) + C ; 8 args: neg_a, A, neg_b, B, c_mod, C, reuse_a, reuse_b
        acc = __builtin_amdgcn_wmma_f32_16x16x4_f32(false, a, false, b,
                                                    (short)0, acc, false, false);
    }

    // C/D layout: VGPR r -> row M = r + 8*hi ; lane -> col N = l16
    #pragma unroll
    for (int r = 0; r < 8; ++r) {
        int row = mbase + r + hi * 8;
        S[(size_t)row * OUT_DIM + n0 + l16] = acc[r];
    }
}

// ---------------------------------------------------------------------------
// Kernel 2: COO SpMM scatter:  out[r] += val * support[c]
//   one wave per edge; lane l handles the float4 at columns [4l, 4l+4)
//   (support and out are L2-resident: 51.2 MB each vs 192 MB L2)
// ---------------------------------------------------------------------------
__global__ void __launch_bounds__(256) gc_spmm(const float* __restrict__ support,
                                               const int*  __restrict__ rows,
                                               const int*  __restrict__ cols,
                                               const float* __restrict__ vals,
                                               float* __restrict__ out,
                                               int nnz) {
    int wave = (blockIdx.x * blockDim.x + threadIdx.x) >> 5;
    int lane = threadIdx.x & 31;
    if (wave >= nnz) return;

    int   r = rows[wave];
    int   c = cols[wave];
    float v = vals[wave];

    float4 g = ((const float4*)(support + (size_t)c * OUT_DIM))[lane];
    float* ob = out + (size_t)r * OUT_DIM + lane * 4;
    atomicAdd(ob + 0, v * g.x);
    atomicAdd(ob + 1, v * g.y);
    atomicAdd(ob + 2, v * g.z);
    atomicAdd(ob + 3, v * g.w);
}

// ---------------------------------------------------------------------------
extern "C" void kernel_launch(void* const* d_in, const int* in_sizes, int n_in,
                              void* d_out, int out_size, void* d_ws, size_t ws_size,
                              hipStream_t stream) {
    const float* X    = (const float*)d_in[0];   // [N, 256]
    const float* W    = (const float*)d_in[1];   // [256, 128]
    const float* bias = (const float*)d_in[2];   // [128]
    const int*   rows = (const int*)d_in[3];     // [nnz]
    const int*   cols = (const int*)d_in[4];     // [nnz]
    const float* vals = (const float*)d_in[5];   // [nnz]
    float* out = (float*)d_out;

    const int nrows = in_sizes[0] / IN_DIM;      // 100000 (divisible by 16)
    const int nnz   = in_sizes[3];               // 3200000
    float* support  = (float*)d_ws;              // [N, 128] f32 scratch (51.2 MB)

    const int total = nrows * OUT_DIM;
    gc_init_bias<<<(total + 255) / 256, 256, 0, stream>>>(bias, out, total);
    gc_gemm_wmma<<<nrows / 16, 256, 0, stream>>>(X, W, support);
    const int edges_per_block = 256 / 32;        // 8 waves per block
    gc_spmm<<<(nnz + edges_per_block - 1) / edges_per_block, 256, 0, stream>>>(
        support, rows, cols, vals, out, nnz);
}